// MBSAAtomCenteredDescriptor_30081950941811
// MI455X (gfx1250) — compile-verified
//
#include <hip/hip_runtime.h>
#include <math.h>

// ---------------- problem constants (match reference) ----------------
#define N_ATOMS   20000
#define N_EDGES   320000
#define FDIM      32
#define LDIM      9
#define NBAS      8
#define NRAD      32
#define CUTOFF_F  5.0f
#define GAMMA_F   (1.0f/1.5f)
#define NODE_FEAT ((long)N_ATOMS * LDIM * FDIM)   // 5,760,000 floats per feature slab
#define NTILES    (N_ATOMS / 16)                  // 1250 16-row tiles
#define GEMM_BLKS ((NTILES + 7) / 8)              // 8 waves (tiles) per 256-thread block

typedef float v2f __attribute__((ext_vector_type(2)));
typedef float v8f __attribute__((ext_vector_type(8)));

// binomial(31,k) and binomial(7,k) for exp-Bernstein bases
__constant__ float COMB31[32] = {
  1.f, 31.f, 465.f, 4495.f, 31465.f, 169911.f, 736281.f, 2629575.f,
  7888725.f, 20160075.f, 44352165.f, 84672315.f, 141120525.f, 206253075.f,
  265182525.f, 300540195.f, 300540195.f, 265182525.f, 206253075.f, 141120525.f,
  84672315.f, 44352165.f, 20160075.f, 7888725.f, 2629575.f, 736281.f,
  169911.f, 31465.f, 4495.f, 465.f, 31.f, 1.f };
__constant__ float COMB7[8] = { 1.f, 7.f, 21.f, 35.f, 35.f, 21.f, 7.f, 1.f };

// ---------------- utility ----------------
__global__ void zero_f32x4_kernel(float4* __restrict__ p, long n4) {
  long i = (long)blockIdx.x * blockDim.x + threadIdx.x;
  long stride = (long)gridDim.x * blockDim.x;
  float4 z = make_float4(0.f, 0.f, 0.f, 0.f);
  for (; i < n4; i += stride) p[i] = z;
}

// ---------------- edge geometry + initial scatter ----------------
// one wave (32 lanes) per edge; lane == radial channel f
__global__ __launch_bounds__(256) void edge_init_kernel(
    const int*   __restrict__ nbr,     // [E][2]
    const float* __restrict__ disp,    // [E][3]
    const int*   __restrict__ Z,       // [N]
    const float* __restrict__ Wsp,     // [100][32]
    float* __restrict__ sh_out,        // [E][9]
    float* __restrict__ rad8_out,      // [E][8]
    float* __restrict__ y)             // [N][9][32] accumulated atomically
{
  int wave = (int)((blockIdx.x * blockDim.x + threadIdx.x) >> 5);
  int lane = threadIdx.x & 31;
  if (wave >= N_EDGES) return;
  int i = nbr[2 * wave + 0];
  int j = nbr[2 * wave + 1];
  float dx = disp[3 * wave + 0], dy = disp[3 * wave + 1], dz = disp[3 * wave + 2];
  float r = sqrtf(dx * dx + dy * dy + dz * dz);
  float inv = 1.0f / (r + 1e-12f);
  float ux = dx * inv, uy = dy * inv, uz = dz * inv;

  float sh[9];
  sh[0] = 0.28209479177387814f;
  sh[1] = 0.4886025119029199f * uy;
  sh[2] = 0.4886025119029199f * uz;
  sh[3] = 0.4886025119029199f * ux;
  sh[4] = 1.0925484305920792f * ux * uy;
  sh[5] = 1.0925484305920792f * uy * uz;
  sh[6] = 0.31539156525252005f * (3.0f * uz * uz - 1.0f);
  sh[7] = 1.0925484305920792f * ux * uz;
  sh[8] = 0.5462742152960396f * (ux * ux - uy * uy);
  if (lane < 9) sh_out[(long)wave * 9 + lane] = sh[lane];

  float fcut = (r < CUTOFF_F) ? 0.5f * (cosf(3.14159265358979323846f * r / CUTOFF_F) + 1.0f) : 0.0f;
  float t = expf(-GAMMA_F * r);
  t = fminf(fmaxf(t, 1e-7f), 1.0f - 1e-7f);
  float omt = 1.0f - t;

  // 32-wide radial basis, modulated by species of source atom j
  float kf = (float)lane;
  float b32 = COMB31[lane] * powf(t, kf) * powf(omt, 31.0f - kf) * fcut;
  float rad = b32 * Wsp[(long)Z[j] * NRAD + lane];

  if (lane < 8) {
    float b8 = COMB7[lane] * powf(t, kf) * powf(omt, 7.0f - kf) * fcut;
    rad8_out[(long)wave * 8 + lane] = b8;
  }

  float* yi = y + (long)i * LDIM * FDIM;
#pragma unroll
  for (int l = 0; l < LDIM; ++l)
    atomicAdd(&yi[l * FDIM + lane], sh[l] * rad);
}

// ---------------- WMMA helpers: C[16x32] (+)= A[16x32] * W[32x32], f32 ----------------
// V_WMMA_F32_16X16X4_F32 fragment layouts per ISA 7.12.2:
//  A 16x4: lane m = lane&15 holds row m; K = 2*(lane>>4) + vgpr
//  B 4x16: vgpr v holds K-row (2*(lane>>4)+v), col = lane&15
//  C 16x16: component r holds row r + 8*(lane>>4), col = lane&15
//
// W is staged in LDS TRANSPOSED (WsT[col][k]) so every B fragment is two
// adjacent floats -> single ds_load_b64 into an even-aligned VGPR pair.
__device__ __forceinline__ void stage_W_transposed(float* WsT, const float* __restrict__ W) {
  for (int t = threadIdx.x; t < 1024; t += 256) {
    int k = t >> 5, c = t & 31;
    WsT[c * 32 + k] = W[t];         // WsT[c][k] = W[k][c]
  }
}

__device__ __forceinline__ void wave_gemm_16x32(
    const float* __restrict__ a_row,   // this lane's A row (row = lane&15), 32 contiguous floats
    const float* WsT,                  // transposed 32x32 weights (LDS)
    v8f& c0, v8f& c1, int lane)
{
  int kh  = lane >> 4;
  int col = lane & 15;
  const float* wc0 = WsT + col * 32;          // column col   (contiguous in k)
  const float* wc1 = WsT + (col + 16) * 32;   // column col+16
#pragma unroll
  for (int kk = 0; kk < 32; kk += 4) {
    int k0 = kk + 2 * kh;                      // even -> 8B aligned
    v2f a  = *(const v2f*)(a_row + k0);
    v2f b0 = *(const v2f*)(wc0 + k0);
    v2f b1 = *(const v2f*)(wc1 + k0);
    c0 = __builtin_amdgcn_wmma_f32_16x16x4_f32(false, a, false, b0, (short)0, c0, false, false);
    c1 = __builtin_amdgcn_wmma_f32_16x16x4_f32(false, a, false, b1, (short)0, c1, false, false);
  }
}

// Generic 16-row GEMM, 8 tiles per 256-thread block: Out[row] (+)= A[gather(row)] @ W
__global__ __launch_bounds__(256) void gemm32_wmma_kernel(
    const float* __restrict__ A, long a_stride,
    const int*   __restrict__ gather,       // optional row gather (e.g. atomic numbers)
    const float* __restrict__ W,            // 32x32
    float* __restrict__ Out, long o_stride,
    int accumulate)
{
  __shared__ float WsT[1024];
  stage_W_transposed(WsT, W);
  __syncthreads();

  int tile = blockIdx.x * 8 + (threadIdx.x >> 5);
  if (tile >= NTILES) return;
  int lane = threadIdx.x & 31;
  int rowbase = tile * 16;
  int arow = rowbase + (lane & 15);
  if (gather) arow = gather[arow];
  const float* a_row = A + (long)arow * a_stride;
  v8f c0 = {}; v8f c1 = {};
  wave_gemm_16x32(a_row, WsT, c0, c1, lane);
  int kh = lane >> 4, col = lane & 15;
#pragma unroll
  for (int r = 0; r < 8; ++r) {
    float* o = Out + (long)(rowbase + r + 8 * kh) * o_stride;
    if (accumulate) { o[col] += c0[r]; o[col + 16] += c1[r]; }
    else            { o[col]  = c0[r]; o[col + 16]  = c1[r]; }
  }
}

// Fused Q/K projection: A rows (s = X[:,0,:]) loaded once, 4 WMMA accumulator chains.
__global__ __launch_bounds__(256) void qk_wmma_kernel(
    const float* __restrict__ X,      // [N][9][32], row 0 = s
    const float* __restrict__ Wq, const float* __restrict__ Wk,
    float* __restrict__ Q, float* __restrict__ K)
{
  __shared__ float WqT[1024];
  __shared__ float WkT[1024];
  stage_W_transposed(WqT, Wq);
  stage_W_transposed(WkT, Wk);
  __syncthreads();

  int tile = blockIdx.x * 8 + (threadIdx.x >> 5);
  if (tile >= NTILES) return;
  int lane = threadIdx.x & 31;
  int rowbase = tile * 16;
  const float* a_row = X + (long)(rowbase + (lane & 15)) * (LDIM * FDIM);

  int kh = lane >> 4, col = lane & 15;
  const float* q_c0 = WqT + col * 32;
  const float* q_c1 = WqT + (col + 16) * 32;
  const float* k_c0 = WkT + col * 32;
  const float* k_c1 = WkT + (col + 16) * 32;
  v8f q0 = {}, q1 = {}, k0v = {}, k1v = {};
#pragma unroll
  for (int kk = 0; kk < 32; kk += 4) {
    int k0 = kk + 2 * kh;
    v2f a   = *(const v2f*)(a_row + k0);
    v2f bq0 = *(const v2f*)(q_c0 + k0);
    v2f bq1 = *(const v2f*)(q_c1 + k0);
    v2f bk0 = *(const v2f*)(k_c0 + k0);
    v2f bk1 = *(const v2f*)(k_c1 + k0);
    q0  = __builtin_amdgcn_wmma_f32_16x16x4_f32(false, a, false, bq0, (short)0, q0,  false, false);
    q1  = __builtin_amdgcn_wmma_f32_16x16x4_f32(false, a, false, bq1, (short)0, q1,  false, false);
    k0v = __builtin_amdgcn_wmma_f32_16x16x4_f32(false, a, false, bk0, (short)0, k0v, false, false);
    k1v = __builtin_amdgcn_wmma_f32_16x16x4_f32(false, a, false, bk1, (short)0, k1v, false, false);
  }
#pragma unroll
  for (int r = 0; r < 8; ++r) {
    long row = rowbase + r + 8 * kh;
    Q[row * FDIM + col] = q0[r];  Q[row * FDIM + col + 16] = q1[r];
    K[row * FDIM + col] = k0v[r]; K[row * FDIM + col + 16] = k1v[r];
  }
}

// Per-degree transform: Out[n,l,:] = X[n,l,:] @ W3[deg(l)]  (used for Wv and Wtd)
__global__ __launch_bounds__(256) void deg_gemm_wmma_kernel(
    const float* __restrict__ X,      // [N][9][32]
    const float* __restrict__ W3,     // [3][32][32]
    float* __restrict__ Out)          // [N][9][32]
{
  __shared__ float WsT[1024];
  int l = blockIdx.y;
  int deg = (l == 0) ? 0 : ((l < 4) ? 1 : 2);   // DEGS = [0,1,1,1,2,2,2,2,2]
  stage_W_transposed(WsT, W3 + (long)deg * 1024);
  __syncthreads();

  int tile = blockIdx.x * 8 + (threadIdx.x >> 5);
  if (tile >= NTILES) return;
  int lane = threadIdx.x & 31;
  int nb = tile * 16;
  const float* a_row = X + ((long)(nb + (lane & 15)) * LDIM + l) * FDIM;
  v8f c0 = {}; v8f c1 = {};
  wave_gemm_16x32(a_row, WsT, c0, c1, lane);
  int kh = lane >> 4, col = lane & 15;
#pragma unroll
  for (int r = 0; r < 8; ++r) {
    float* o = Out + ((long)(nb + r + 8 * kh) * LDIM + l) * FDIM;
    o[col] = c0[r]; o[col + 16] = c1[r];
  }
}

// ---------------- attention edge kernels ----------------
// score + segment max (order-preserving unsigned encoding for float atomicMax)
__global__ __launch_bounds__(256) void score_max_kernel(
    const int*   __restrict__ nbr,
    const float* __restrict__ Q,     // [N][32]
    const float* __restrict__ K,     // [N][32]
    float*    __restrict__ score,    // [E]
    unsigned* __restrict__ smax)     // [N] encoded, init 0
{
  int wave = (int)((blockIdx.x * blockDim.x + threadIdx.x) >> 5);
  int lane = threadIdx.x & 31;
  if (wave >= N_EDGES) return;
  int i = nbr[2 * wave + 0];
  int j = nbr[2 * wave + 1];
  float p = Q[(long)i * FDIM + lane] * K[(long)j * FDIM + lane];
#pragma unroll
  for (int off = 16; off; off >>= 1) p += __shfl_xor(p, off, 32);
  float s = p * 0.17677669529663687f;   // 1/sqrt(32)
  if (lane == 0) {
    score[wave] = s;
    unsigned bits = __float_as_uint(s);
    unsigned enc = (bits & 0x80000000u) ? ~bits : (bits | 0x80000000u);
    atomicMax(&smax[i], enc);
  }
}

__global__ __launch_bounds__(256) void exp_denom_kernel(
    const int*      __restrict__ nbr,
    const float*    __restrict__ score,
    const unsigned* __restrict__ smax,
    float* __restrict__ esc,          // [E]
    float* __restrict__ denom)        // [N], init 0
{
  int e = blockIdx.x * blockDim.x + threadIdx.x;
  if (e >= N_EDGES) return;
  int i = nbr[2 * e + 0];
  unsigned enc = smax[i];
  unsigned bits = (enc & 0x80000000u) ? (enc ^ 0x80000000u) : ~enc;
  float mx = __uint_as_float(bits);
  float ev = expf(score[e] - mx);
  esc[e] = ev;
  atomicAdd(&denom[i], ev);
}

// msg[e,l,f] = v[j,l,f] * sh[e,l] * (rad8[e]@Wb)[f] * alpha[e]  -> segment_sum into out[i]
__global__ __launch_bounds__(256) void msg_scatter_kernel(
    const int*   __restrict__ nbr,
    const float* __restrict__ vbuf,   // [N][9][32]
    const float* __restrict__ sh,     // [E][9]
    const float* __restrict__ rad8,   // [E][8]
    const float* __restrict__ esc,    // [E]
    const float* __restrict__ denom,  // [N]
    const float* __restrict__ Wb,     // [8][32]
    float* __restrict__ outx)         // [N][9][32], init 0
{
  int wave = (int)((blockIdx.x * blockDim.x + threadIdx.x) >> 5);
  int lane = threadIdx.x & 31;
  if (wave >= N_EDGES) return;
  int i = nbr[2 * wave + 0];
  int j = nbr[2 * wave + 1];
  float alpha = esc[wave] / (denom[i] + 1e-12f);

  const float* r8 = rad8 + (long)wave * 8;
  float rw = 0.0f;
#pragma unroll
  for (int k = 0; k < NBAS; ++k) rw = fmaf(r8[k], Wb[k * FDIM + lane], rw);
  rw *= alpha;

  const float* vj  = vbuf + (long)j * LDIM * FDIM;
  const float* she = sh   + (long)wave * LDIM;
  float* oi = outx + (long)i * LDIM * FDIM;
#pragma unroll
  for (int l = 0; l < LDIM; ++l)
    atomicAdd(&oi[l * FDIM + lane], vj[l * FDIM + lane] * she[l] * rw);
}

// ---------------- host-side orchestration ----------------
static void launch_zero(float* p, long n, hipStream_t s) {   // n must be multiple of 4
  long n4 = n >> 2;
  int blocks = (int)((n4 + 255) / 256);
  if (blocks > 2048) blocks = 2048;
  zero_f32x4_kernel<<<blocks, 256, 0, s>>>((float4*)p, n4);
}

static void attn_step(const int* nbr,
                      const float* cur, float* tgt, float* vbuf,
                      const float* sh, const float* rad8,
                      float* Q, float* K, float* score, float* esc,
                      unsigned* smax, float* denom,
                      const float* Wq, const float* Wk,
                      const float* Wv, const float* Wb,
                      hipStream_t stream)
{
  const int EW = (N_EDGES * 32 + 255) / 256; // wave-per-edge grid
  qk_wmma_kernel<<<GEMM_BLKS, 256, 0, stream>>>(cur, Wq, Wk, Q, K);
  launch_zero((float*)smax, 2 * N_ATOMS, stream);  // smax (0 == encoded -inf) + denom, contiguous
  launch_zero(tgt, NODE_FEAT, stream);
  score_max_kernel<<<EW, 256, 0, stream>>>(nbr, Q, K, score, smax);
  exp_denom_kernel<<<(N_EDGES + 255) / 256, 256, 0, stream>>>(nbr, score, smax, esc, denom);
  deg_gemm_wmma_kernel<<<dim3(GEMM_BLKS, LDIM), 256, 0, stream>>>(cur, Wv, vbuf);
  msg_scatter_kernel<<<EW, 256, 0, stream>>>(nbr, vbuf, sh, rad8, esc, denom, Wb, tgt);
}

extern "C" void kernel_launch(void* const* d_in, const int* in_sizes, int n_in,
                              void* d_out, int out_size, void* d_ws, size_t ws_size,
                              hipStream_t stream) {
  (void)in_sizes; (void)n_in; (void)out_size; (void)ws_size;
  // JAX pytree (sorted-key) flattening order of setup_inputs():
  const int*   Z    = (const int*)d_in[0];                  // atomic_numbers [N]
  const int*   nbr  = (const int*)d_in[1];                  // neighbour_indices [E][2]
  const float* disp = (const float*)d_in[2];                // displacements [E][3]
  const float* W_et = (const float*)d_in[3];                // [32][32]
  const float* Wsp  = (const float*)d_in[4];                // W_species [100][32]
  const float* Wtd  = (const float*)d_in[5];                // [3][32][32]
  const float *Wb[4], *Wk[4], *Wq[4], *Wv[4];
  for (int a = 0; a < 4; ++a) {                             // attn dicts: Wb,Wk,Wq,Wv sorted
    Wb[a] = (const float*)d_in[6 + 4 * a];
    Wk[a] = (const float*)d_in[7 + 4 * a];
    Wq[a] = (const float*)d_in[8 + 4 * a];
    Wv[a] = (const float*)d_in[9 + 4 * a];
  }
  const float* embed = (const float*)d_in[22];              // [100][32]

  float* out0 = (float*)d_out;
  float* out1 = out0 + NODE_FEAT;

  float* ws    = (float*)d_ws;
  float* xA    = ws;
  float* xB    = xA + NODE_FEAT;
  float* vbuf  = xB + NODE_FEAT;
  float* sh    = vbuf + NODE_FEAT;
  float* rad8  = sh + (long)N_EDGES * LDIM;
  float* Q     = rad8 + (long)N_EDGES * NBAS;
  float* K     = Q + (long)N_ATOMS * FDIM;
  float* score = K + (long)N_ATOMS * FDIM;
  float* esc   = score + N_EDGES;
  unsigned* smax = (unsigned*)(esc + N_EDGES);
  float* denom = (float*)(smax + N_ATOMS);

  const int EW = (N_EDGES * 32 + 255) / 256;

  // y = segment_sum(sh (x) rad32*W_species) ; y[:,0,:] += embed[Z] @ W_et
  launch_zero(xA, NODE_FEAT, stream);
  edge_init_kernel<<<EW, 256, 0, stream>>>(nbr, disp, Z, Wsp, sh, rad8, xA);
  gemm32_wmma_kernel<<<GEMM_BLKS, 256, 0, stream>>>(embed, NRAD, Z, W_et, xA, LDIM * FDIM, 1);

  // block 0: two attention steps; second writes straight into output slab 0
  attn_step(nbr, xA, xB,   vbuf, sh, rad8, Q, K, score, esc, smax, denom,
            Wq[0], Wk[0], Wv[0], Wb[0], stream);
  attn_step(nbr, xB, out0, vbuf, sh, rad8, Q, K, score, esc, smax, denom,
            Wq[1], Wk[1], Wv[1], Wb[1], stream);

  // inter-block per-degree mix (Wtd) via WMMA
  deg_gemm_wmma_kernel<<<dim3(GEMM_BLKS, LDIM), 256, 0, stream>>>(out0, Wtd, xA);

  // block 1: two attention steps; second writes into output slab 1
  attn_step(nbr, xA, xB,   vbuf, sh, rad8, Q, K, score, esc, smax, denom,
            Wq[2], Wk[2], Wv[2], Wb[2], stream);
  attn_step(nbr, xB, out1, vbuf, sh, rad8, Q, K, score, esc, smax, denom,
            Wq[3], Wk[3], Wv[3], Wb[3], stream);
}